// GriddingDistance_55224689492786
// MI455X (gfx1250) — compile-verified
//
#include <hip/hip_runtime.h>

// GriddingDistance forward for MI455X (gfx1250, wave32).
// Strategy: zero-fill the 128 MiB output (fits in 192 MB L2), then one thread
// per point issues 8 non-returning global_atomic_add_f32 ops that resolve in L2.

namespace {
constexpr int   R     = 128;
constexpr int   G     = R * R * R;     // 2,097,152 voxels per sample
constexpr int   B     = 8;             // batch
constexpr int   NPTS  = 65536;         // points per sample (= 1 << 16)
constexpr float SCALE = 128.0f;        // GriddingDistance scale
constexpr float SHIFT = 64.0f;         // -GRID_MIN
}

__global__ __launch_bounds__(256) void gd_zero_kernel(float4* __restrict__ out) {
    // one b128 store per thread; exact-sized launch
    unsigned i = blockIdx.x * 256u + threadIdx.x;
    out[i] = make_float4(0.f, 0.f, 0.f, 0.f);
}

__device__ __forceinline__ void atomic_add_agent(float* p, float v) {
    // relaxed + agent scope, result unused -> non-returning global_atomic_add_f32
    (void)__hip_atomic_fetch_add(p, v, __ATOMIC_RELAXED, __HIP_MEMORY_SCOPE_AGENT);
}

__global__ __launch_bounds__(256) void gd_scatter_kernel(const float* __restrict__ cloud,
                                                         float* __restrict__ grid) {
    const unsigned tid = blockIdx.x * 256u + threadIdx.x;   // [0, B*NPTS)
    const unsigned b   = tid >> 16;                         // NPTS == 1<<16

    // packed xyz per point (12 B); wave-contiguous so loads coalesce
    const float px = cloud[3u * tid + 0] * SCALE + SHIFT;
    const float py = cloud[3u * tid + 1] * SCALE + SHIFT;
    const float pz = cloud[3u * tid + 2] * SCALE + SHIFT;

    const float fx = floorf(px), fy = floorf(py), fz = floorf(pz);
    const float dx = px - fx,    dy = py - fy,    dz = pz - fz;

    const int x0 = (int)fx, y0 = (int)fy, z0 = (int)fz;
    // clip(x0 + corner, 0, R-1) — branch-free, matches reference semantics
    const int x0c = min(max(x0,     0), R - 1);
    const int x1c = min(max(x0 + 1, 0), R - 1);
    const int y0c = min(max(y0,     0), R - 1);
    const int y1c = min(max(y0 + 1, 0), R - 1);
    const int z0c = min(max(z0,     0), R - 1);
    const int z1c = min(max(z0 + 1, 0), R - 1);

    const float wx0 = 1.0f - dx, wx1 = dx;
    const float wy0 = 1.0f - dy, wy1 = dy;
    const float wz0 = 1.0f - dz, wz1 = dz;

    float* g = grid + (size_t)b * (size_t)G;
    const int bx0 = x0c * (R * R), bx1 = x1c * (R * R);
    const int by0 = y0c * R,       by1 = y1c * R;

    // 8 corners; z-pair is contiguous in the flat grid (good L2 sector reuse)
    atomic_add_agent(&g[bx0 + by0 + z0c], wx0 * wy0 * wz0);
    atomic_add_agent(&g[bx0 + by0 + z1c], wx0 * wy0 * wz1);
    atomic_add_agent(&g[bx0 + by1 + z0c], wx0 * wy1 * wz0);
    atomic_add_agent(&g[bx0 + by1 + z1c], wx0 * wy1 * wz1);
    atomic_add_agent(&g[bx1 + by0 + z0c], wx1 * wy0 * wz0);
    atomic_add_agent(&g[bx1 + by0 + z1c], wx1 * wy0 * wz1);
    atomic_add_agent(&g[bx1 + by1 + z0c], wx1 * wy1 * wz0);
    atomic_add_agent(&g[bx1 + by1 + z1c], wx1 * wy1 * wz1);
}

extern "C" void kernel_launch(void* const* d_in, const int* in_sizes, int n_in,
                              void* d_out, int out_size, void* d_ws, size_t ws_size,
                              hipStream_t stream) {
    (void)in_sizes; (void)n_in; (void)d_ws; (void)ws_size;

    const float* pred = (const float*)d_in[0];   // (8, 65536, 3) fp32
    const float* gt   = (const float*)d_in[1];   // (8, 65536, 3) fp32
    float*       out  = (float*)d_out;           // pred_grid (8,G) ++ gt_grid (8,G)

    // 1) zero the full output (2 * B * G floats = out_size); establishes grid in L2
    const int n4 = out_size / 4;                 // float4 stores
    gd_zero_kernel<<<n4 / 256, 256, 0, stream>>>((float4*)out);

    // 2) scatter both clouds (same stream -> ordered after zero-fill)
    const int total_pts = B * NPTS;              // 524,288 threads, one per point
    gd_scatter_kernel<<<total_pts / 256, 256, 0, stream>>>(pred, out);
    gd_scatter_kernel<<<total_pts / 256, 256, 0, stream>>>(gt,   out + (size_t)B * (size_t)G);
}